// Actor_67422396612916
// MI455X (gfx1250) — compile-verified
//
#include <hip/hip_runtime.h>
#include <hip/hip_bf16.h>
#include <math.h>

// Problem constants (match reference)
#define BB      16
#define NN      1024
#define ND      32
#define GC1     64
#define GC2     32
#define AUXD    128
#define LIND    128
#define ZD      16
#define NCHUNK  16          // N / 64 node-chunks for gate partial sums

typedef float v2f __attribute__((ext_vector_type(2)));
typedef float v8f __attribute__((ext_vector_type(8)));
typedef int   v4i __attribute__((ext_vector_type(4)));

// ---- CDNA5 async global->LDS copy (no VGPR data residency; ASYNCcnt) ------
#if __has_builtin(__builtin_amdgcn_global_load_async_to_lds_b128) && \
    __has_builtin(__builtin_amdgcn_s_wait_asynccnt)
#define HAVE_ASYNC_LDS 1
#else
#define HAVE_ASYNC_LDS 0
#endif

__device__ __forceinline__ void async_cp16(const float* g, float* l) {
#if HAVE_ASYNC_LDS
    __builtin_amdgcn_global_load_async_to_lds_b128(
        (__attribute__((address_space(1))) v4i*)(void*)g,
        (__attribute__((address_space(3))) v4i*)(void*)l, 0, 0);
#else
    *(float4*)l = *(const float4*)g;
#endif
}
__device__ __forceinline__ void async_wait0() {
#if HAVE_ASYNC_LDS
    __builtin_amdgcn_s_wait_asynccnt(0);
#endif
}

// ---------------------------------------------------------------------------
// X1 = node @ W1 + b1   (one thread per output element; tiny GEMM)
// ---------------------------------------------------------------------------
__global__ void k_linear(const float* __restrict__ X, const float* __restrict__ W,
                         const float* __restrict__ bias, float* __restrict__ Y,
                         int K, int C, int total) {
    int idx = blockIdx.x * blockDim.x + threadIdx.x;
    if (idx >= total) return;
    int r = idx / C, c = idx % C;
    float s = bias[c];
    const float* xr = X + (size_t)r * K;
    for (int k = 0; k < K; ++k) s = fmaf(xr[k], W[k * C + c], s);
    Y[idx] = s;
}

// ---------------------------------------------------------------------------
// h2 = (S1 + X1) @ W2 + b2
// ---------------------------------------------------------------------------
__global__ void k_lin2(const float* __restrict__ S1, const float* __restrict__ X1,
                       const float* __restrict__ W2, const float* __restrict__ b2,
                       float* __restrict__ h2, int total) {
    int idx = blockIdx.x * blockDim.x + threadIdx.x;
    if (idx >= total) return;
    int r = idx / GC2, c = idx % GC2;
    float s = b2[c];
    const float* s1 = S1 + (size_t)r * GC1;
    const float* x1 = X1 + (size_t)r * GC1;
    for (int k = 0; k < GC1; ++k) s = fmaf(s1[k] + x1[k], W2[k * GC2 + c], s);
    h2[idx] = s;
}

// ---------------------------------------------------------------------------
// WMMA f32 SpMM:  S = adj_s @ X, 16-row x C-col tile per block, one wave per
// 16 output columns. MODE: 0 = sum raw channels (NT stream), 1 = sum raw +
// cache adj_s to global (RT -> stays in 192MB L2), 2 = read cached adj_s.
// LDS double buffer; X tile staged with async global->LDS b128 copies so
// no data-carrying registers live across the WMMA block (no spills).
// ---------------------------------------------------------------------------
template <int C, int MODE>
__global__ __launch_bounds__(32 * (C / 16))
void k_spmm(const float* __restrict__ adj_raw,   // (B,N,N,5)
            const float* __restrict__ adj_sum,   // (B,N,N) pre-summed
            float* __restrict__ adj_sum_out,     // cache target
            const float* __restrict__ X,         // (B*N, C)
            float* __restrict__ S)               // (B*N, C)
{
    constexpr int T  = 32 * (C / 16);    // threads per block
    constexpr int KC = 32;               // K-chunk
    constexpr int AS = 36;               // As row stride (16B rows, no b64 conflicts)
    constexpr int XS = C + 8;            // Xs row stride (16B rows, 2*XS%64==16)
    constexpr int AE = (16 * KC) / T;       // A floats per thread (MODE 0/1)
    constexpr int AQ = (16 * KC) / (4 * T); // A float4 asyncs per thread (MODE 2)
    constexpr int XQ = (KC * C) / (4 * T);  // X float4 asyncs per thread (==4)

    __shared__ __align__(16) float As[2][16 * AS];
    __shared__ __align__(16) float Xs[2][KC * XS];

    const int b    = blockIdx.y;
    const int row0 = blockIdx.x * 16;
    const int tid  = threadIdx.x;
    const int wave = tid >> 5;
    const int lane = tid & 31;
    const int m    = lane & 15;
    const int half = lane >> 4;
    const int col0 = wave * 16;
    const size_t rowbase = (size_t)(b * NN + row0) * NN;   // A tile row origin

    auto issue_X = [&](int k0, int buf) {
        const float* gp = X + (size_t)(b * NN + k0) * C;
#pragma unroll
        for (int i = 0; i < XQ; ++i) {
            int q = tid + i * T;               // float4 index within tile
            int f = q * 4;
            int rr = f / C, cc = f % C;
            async_cp16(gp + f, &Xs[buf][rr * XS + cc]);
        }
    };
    auto issue_A_async = [&](int k0, int buf) {        // MODE 2
#pragma unroll
        for (int i = 0; i < AQ; ++i) {
            int q  = tid + i * T;
            int mm = q >> 3, k4 = (q & 7) * 4;
            async_cp16(adj_sum + rowbase + (size_t)mm * NN + (size_t)(k0 + k4),
                       &As[buf][mm * AS + k4]);
        }
    };
    auto load_A_regs = [&](int k0, float* avr) {       // MODE 0/1 (NT stream)
#pragma unroll
        for (int i = 0; i < AE; ++i) {
            int e  = tid + i * T;
            int mm = e >> 5, kk = e & 31;
            const float* p = adj_raw + (rowbase + (size_t)mm * NN + (size_t)(k0 + kk)) * 5;
            avr[i] = __builtin_nontemporal_load(p + 1) +
                     __builtin_nontemporal_load(p + 2) +
                     __builtin_nontemporal_load(p + 3) +
                     __builtin_nontemporal_load(p + 4);
        }
    };
    auto store_A_regs = [&](int k0, const float* avr, int buf) {
#pragma unroll
        for (int i = 0; i < AE; ++i) {
            int e  = tid + i * T;
            int mm = e >> 5, kk = e & 31;
            As[buf][mm * AS + kk] = avr[i];
            if (MODE == 1)
                adj_sum_out[rowbase + (size_t)mm * NN + (size_t)(k0 + kk)] = avr[i];
        }
    };

    float av[AE];
    // ---- prologue: stage chunk 0 into buffer 0 ----
    issue_X(0, 0);
    if (MODE == 2) issue_A_async(0, 0);
    else           load_A_regs(0, av);

    v8f acc = {};
    for (int k0 = 0; k0 < NN; k0 += KC) {
        const int buf = (k0 >> 5) & 1;
        if (MODE != 2) store_A_regs(k0, av, buf);
        async_wait0();           // own async writes to buf have landed in LDS
        __syncthreads();         // buf visible to all; all done reading buf^1

        if (k0 + KC < NN) {      // stage chunk i+1 under the WMMAs
            issue_X(k0 + KC, buf ^ 1);
            if (MODE == 2) issue_A_async(k0 + KC, buf ^ 1);
            else           load_A_regs(k0 + KC, av);
        }

        // ---- 8 x WMMA f32 16x16x4 ----
#pragma unroll
        for (int kk = 0; kk < KC; kk += 4) {
            const int ka = kk + 2 * half;              // even -> 8B aligned
            v2f a = *(const v2f*)&As[buf][m * AS + ka];
            v2f bb;
            bb.x = Xs[buf][ka * XS + col0 + m];
            bb.y = Xs[buf][(ka + 1) * XS + col0 + m];
            acc = __builtin_amdgcn_wmma_f32_16x16x4_f32(
                false, a, false, bb, (short)0, acc, false, false);
        }
    }

    // ---- store 16x16 tile: VGPR r -> row r (lanes 0-15) / r+8 (lanes 16-31)
#pragma unroll
    for (int r = 0; r < 8; ++r) {
        S[((size_t)(b * NN + row0 + r + 8 * half)) * C + col0 + m] = acc[r];
    }
}

// ---------------------------------------------------------------------------
// Gated aggregation: per (batch, 64-node chunk) partial sums of
//   sigmoid(x@Ws+bs) * tanh(x@Wt+bt), x = [S2+h2, node]  (fixed order)
// ---------------------------------------------------------------------------
__global__ void k_gate(const float* __restrict__ S2, const float* __restrict__ h2,
                       const float* __restrict__ node,
                       const float* __restrict__ Ws, const float* __restrict__ bs,
                       const float* __restrict__ Wt, const float* __restrict__ bt,
                       float* __restrict__ pg) {
    const int chunk = blockIdx.x;     // 0..NCHUNK-1 (64 nodes each)
    const int b     = blockIdx.y;
    const int j     = threadIdx.x;    // 0..127
    __shared__ float xv[GC2 + ND];    // 64

    float accum = 0.0f;
    const int row0 = chunk * 64;
    for (int n = 0; n < 64; ++n) {
        const int row = b * NN + row0 + n;
        if (j < GC2 + ND) {
            float v;
            if (j < GC2) v = S2[(size_t)row * GC2 + j] + h2[(size_t)row * GC2 + j];
            else         v = node[(size_t)row * ND + (j - GC2)];
            xv[j] = v;
        }
        __syncthreads();
        float s = bs[j], t = bt[j];
#pragma unroll 8
        for (int k = 0; k < GC2 + ND; ++k) {
            float xk = xv[k];
            s = fmaf(xk, Ws[k * AUXD + j], s);
            t = fmaf(xk, Wt[k * AUXD + j], t);
        }
        float sig = 1.0f / (1.0f + expf(-s));
        accum = fmaf(sig, tanhf(t), accum);
        __syncthreads();
    }
    pg[((size_t)b * NCHUNK + chunk) * AUXD + j] = accum;
}

// ---------------------------------------------------------------------------
// Head: reduce chunks, tanh, 2-layer tanh MLP, final linear. One block/batch.
// ---------------------------------------------------------------------------
__global__ void k_head(const float* __restrict__ pg,
                       const float* __restrict__ Wm1, const float* __restrict__ bm1,
                       const float* __restrict__ Wm2, const float* __restrict__ bm2,
                       const float* __restrict__ Wl,  const float* __restrict__ bl,
                       float* __restrict__ out) {
    const int b = blockIdx.x;
    const int j = threadIdx.x;        // 0..127
    __shared__ float buf[2][AUXD];

    float s = 0.0f;
    for (int c = 0; c < NCHUNK; ++c) s += pg[((size_t)b * NCHUNK + c) * AUXD + j];
    buf[0][j] = tanhf(s);
    __syncthreads();

    float a = bm1[j];
    for (int k = 0; k < AUXD; ++k) a = fmaf(buf[0][k], Wm1[k * LIND + j], a);
    buf[1][j] = tanhf(a);
    __syncthreads();

    a = bm2[j];
    for (int k = 0; k < LIND; ++k) a = fmaf(buf[1][k], Wm2[k * LIND + j], a);
    buf[0][j] = tanhf(a);
    __syncthreads();

    if (j < ZD) {
        float o = bl[j];
        for (int k = 0; k < LIND; ++k) o = fmaf(buf[0][k], Wl[k * ZD + j], o);
        out[(size_t)b * ZD + j] = o;
    }
}

// ---------------------------------------------------------------------------
extern "C" void kernel_launch(void* const* d_in, const int* in_sizes, int n_in,
                              void* d_out, int out_size, void* d_ws, size_t ws_size,
                              hipStream_t stream) {
    const float* node    = (const float*)d_in[0];
    const float* adj_raw = (const float*)d_in[1];
    const float* W1 = (const float*)d_in[2];
    const float* b1 = (const float*)d_in[3];
    const float* W2 = (const float*)d_in[4];
    const float* b2 = (const float*)d_in[5];
    const float* Ws = (const float*)d_in[6];
    const float* bs = (const float*)d_in[7];
    const float* Wt = (const float*)d_in[8];
    const float* bt = (const float*)d_in[9];
    const float* Wm1 = (const float*)d_in[10];
    const float* bm1 = (const float*)d_in[11];
    const float* Wm2 = (const float*)d_in[12];
    const float* bm2 = (const float*)d_in[13];
    const float* Wl  = (const float*)d_in[14];
    const float* bl  = (const float*)d_in[15];
    float* out = (float*)d_out;

    const int ROWS = BB * NN;                 // 16384
    float* ws = (float*)d_ws;
    size_t o = 0;
    float* X1 = ws + o; o += (size_t)ROWS * GC1;     // 1M floats
    float* S1 = ws + o; o += (size_t)ROWS * GC1;     // 1M
    float* H2 = ws + o; o += (size_t)ROWS * GC2;     // 512K
    float* S2 = ws + o; o += (size_t)ROWS * GC2;     // 512K
    float* PG = ws + o; o += (size_t)BB * NCHUNK * AUXD;
    const size_t adjs_elems = (size_t)BB * NN * NN;  // 16.7M floats (67MB)
    float* ADJS = nullptr;
    if (ws_size >= (o + adjs_elems) * sizeof(float)) ADJS = ws + o;

    // 1) X1 = node @ W1 + b1
    {
        int total = ROWS * GC1;
        k_linear<<<(total + 255) / 256, 256, 0, stream>>>(node, W1, b1, X1, ND, GC1, total);
    }
    // 2) S1 = adj_s @ X1   (NT-stream raw channels; cache adj_s in L2-resident ws)
    if (ADJS)
        k_spmm<GC1, 1><<<dim3(NN / 16, BB), 128, 0, stream>>>(adj_raw, nullptr, ADJS, X1, S1);
    else
        k_spmm<GC1, 0><<<dim3(NN / 16, BB), 128, 0, stream>>>(adj_raw, nullptr, nullptr, X1, S1);
    // 3) H2 = (S1 + X1) @ W2 + b2
    {
        int total = ROWS * GC2;
        k_lin2<<<(total + 255) / 256, 256, 0, stream>>>(S1, X1, W2, b2, H2, total);
    }
    // 4) S2 = adj_s @ H2   (cached adj_s if present, else resum raw)
    if (ADJS)
        k_spmm<GC2, 2><<<dim3(NN / 16, BB), 64, 0, stream>>>(nullptr, ADJS, nullptr, H2, S2);
    else
        k_spmm<GC2, 0><<<dim3(NN / 16, BB), 64, 0, stream>>>(adj_raw, nullptr, nullptr, H2, S2);
    // 5) gated aggregation partial sums
    k_gate<<<dim3(NCHUNK, BB), AUXD, 0, stream>>>(S2, H2, node, Ws, bs, Wt, bt, PG);
    // 6) reduce + MLP head
    k_head<<<BB, AUXD, 0, stream>>>(PG, Wm1, bm1, Wm2, bm2, Wl, bl, out);
}